// SwAVLoss_13400297963812
// MI455X (gfx1250) — compile-verified
//
#include <hip/hip_runtime.h>

#define B_N 4096
#define K_N 3000
#define K_PAD 3072
#define C_N 8
#define ISPLIT 16
#define SEG (B_N / ISPLIT)
#define KSPLIT 8
#define NCHUNK (K_N / 8)          // 375 chunks of 8 floats
#define MAXV 384                  // max per-split vec slice (47 chunks * 8, padded)

constexpr float kEps = 0.05f;
constexpr float kTemp = 0.1f;
constexpr int kMinIters = 3;
constexpr float kTol = 1e-3f;
constexpr int kMaxIters = 100;
constexpr float kTiny = 1e-12f;
constexpr float kTargetC = (float)B_N / (float)K_N;

typedef __attribute__((ext_vector_type(2))) float v2f;
typedef __attribute__((ext_vector_type(8))) float v8f;

struct Ptrs8 { const float* p[8]; };
struct Ctrl { int done; int it; unsigned int row_err; unsigned int col_err; };

__device__ __forceinline__ float blkSum(float v, float* red) {
  red[threadIdx.x] = v; __syncthreads();
  for (int o = 128; o > 0; o >>= 1) {
    if ((int)threadIdx.x < o) red[threadIdx.x] += red[threadIdx.x + o];
    __syncthreads();
  }
  float r = red[0]; __syncthreads();
  return r;
}

__device__ __forceinline__ float blkMax(float v, float* red) {
  red[threadIdx.x] = v; __syncthreads();
  for (int o = 128; o > 0; o >>= 1) {
    if ((int)threadIdx.x < o) red[threadIdx.x] = fmaxf(red[threadIdx.x], red[threadIdx.x + o]);
    __syncthreads();
  }
  float r = red[0]; __syncthreads();
  return r;
}

// Q_ij = exp(x_ij/EPS - rowmax(x/EPS)); one block per row.
__global__ __launch_bounds__(256) void buildQ(Ptrs8 ptrs, const int* __restrict__ idxp,
                                              float* __restrict__ Q) {
  __shared__ float xs[K_PAD];
  __shared__ float red[256];
  const int row = blockIdx.x;
  const float* x = ptrs.p[(*idxp) & 7] + (size_t)row * K_N;
  float m = -3.4e38f;
  for (int j = threadIdx.x; j < K_N; j += 256) {
    const float v = x[j] * (1.0f / kEps);
    xs[j] = v;
    m = fmaxf(m, v);
  }
  m = blkMax(m, red);
  float* qr = Q + (size_t)row * K_N;
  for (int j = threadIdx.x; j < K_N; j += 256) qr[j] = expf(xs[j] - m);
}

__global__ void sinkInit(float* __restrict__ u, float* __restrict__ v, Ctrl* c) {
  const int t = blockIdx.x * 256 + threadIdx.x;
  if (t < B_N) u[t] = 1.0f;
  if (t < K_PAD) v[t] = 1.0f;
  if (t == 0) { c->done = 0; c->it = 0; c->row_err = 0u; c->col_err = 0u; }
}

// Partial row reduction over one K-slice: rpart[split][row] = sum_{j in slice} Q_ij*vec_j.
// Each wave: 16 rows. Lane L(<16) covers chunk offsets [0,4), lane L+16 covers [4,8).
// Dual accumulators map to A-fragment slots 0/1; the single
// V_WMMA_F32_16X16X4_F32 against an all-ones B collapses the 16x4 partial
// fragment into 16 exact fp32 row sums (D[m][n] = sum_k A[m][k]).
__global__ __launch_bounds__(256) void rowPass(
    const float* __restrict__ Q, const float* __restrict__ vec,
    float* __restrict__ rpart, const Ctrl* __restrict__ ctrl, int guard) {
  if (guard && ctrl->done) return;
  const int split = blockIdx.y;
  const int cb = (NCHUNK * split) / KSPLIT;
  const int ce = (NCHUNK * (split + 1)) / KSPLIT;
  const int kb = cb * 8, ke = ce * 8;
  __shared__ __align__(16) float vs[MAXV];
  for (int j = threadIdx.x; j < ke - kb; j += 256) vs[j] = vec[kb + j];
  __syncthreads();
  const int wave = threadIdx.x >> 5;
  const int lane = threadIdx.x & 31;
  const int lrow = lane & 15;
  const int hi = lane >> 4;
  const int rbase = blockIdx.x * 128 + wave * 16;
  const float* qrow = Q + (size_t)(rbase + lrow) * K_N + hi * 4;
  float acc0 = 0.0f, acc1 = 0.0f;
  int k0 = kb;
  for (; k0 + 16 <= ke; k0 += 16) {
    const float4 q0 = *(const float4*)(qrow + k0);
    const float4 w0 = *(const float4*)(vs + (k0 - kb) + hi * 4);
    const float4 q1 = *(const float4*)(qrow + k0 + 8);
    const float4 w1 = *(const float4*)(vs + (k0 - kb) + 8 + hi * 4);
    acc0 += q0.x * w0.x + q0.y * w0.y + q0.z * w0.z + q0.w * w0.w;
    acc1 += q1.x * w1.x + q1.y * w1.y + q1.z * w1.z + q1.w * w1.w;
  }
  if (k0 < ke) {
    const float4 q0 = *(const float4*)(qrow + k0);
    const float4 w0 = *(const float4*)(vs + (k0 - kb) + hi * 4);
    acc0 += q0.x * w0.x + q0.y * w0.y + q0.z * w0.z + q0.w * w0.w;
  }
  v2f a; a[0] = acc0; a[1] = acc1;
  v2f b; b[0] = 1.0f; b[1] = 1.0f;   // all-ones B: D[m][n] = rowsum_m
  v8f c = {};
  v8f d = __builtin_amdgcn_wmma_f32_16x16x4_f32(false, a, false, b, (short)0, c,
                                                false, false);
  if (lrow == 0) {                   // lanes 0 & 16 hold rows 0-7 / 8-15 (col 0)
    float* rp = rpart + (size_t)split * B_N + rbase + hi * 8;
#pragma unroll
    for (int g = 0; g < 8; ++g) rp[g] = d[g];
  }
}

// Combine K-split partials in fixed order, then mode epilogue:
// mode 0: out[row] = 1/(r+TINY)                      (u update)
// mode 1: atomicMax(row_err, |uin[row]*r - 1|)        (convergence error)
// mode 2: out[row] = uin[row]/max(uin[row]*r, TINY)   (row renorm factor)
__global__ __launch_bounds__(256) void rowCombine(
    const float* __restrict__ rpart, const float* __restrict__ uin,
    float* __restrict__ out, Ctrl* ctrl, int mode, int guard) {
  if (guard && ctrl->done) return;
  const int row = blockIdx.x * 256 + threadIdx.x;
  if (row >= B_N) return;
  float r = 0.0f;
#pragma unroll
  for (int s = 0; s < KSPLIT; ++s) r += rpart[(size_t)s * B_N + row];
  if (mode == 0) {
    out[row] = 1.0f / (r + kTiny);
  } else if (mode == 1) {
    const float e = fabsf(uin[row] * r - 1.0f);
    atomicMax(&ctrl->row_err, __float_as_uint(e));
  } else {
    const float u = uin[row];
    out[row] = u / fmaxf(u * r, kTiny);
  }
}

// Partial column sums t_j = sum_{i in segment} win_i * Q_ij (coalesced along j).
__global__ __launch_bounds__(256) void colPass(
    const float* __restrict__ Q, const float* __restrict__ win,
    float* __restrict__ tpart, const Ctrl* __restrict__ ctrl, int guard) {
  if (guard && ctrl->done) return;
  __shared__ float wsh[SEG];
  const int seg = blockIdx.y;
  for (int i = threadIdx.x; i < SEG; i += 256) wsh[i] = win[seg * SEG + i];
  __syncthreads();
  const int j = blockIdx.x * 256 + threadIdx.x;
  if (j >= K_N) return;
  const float* qp = Q + (size_t)seg * SEG * K_N + j;
  float t = 0.0f;
  for (int i = 0; i < SEG; ++i) t += wsh[i] * qp[(size_t)i * K_N];
  tpart[seg * K_PAD + j] = t;
}

// mode 0: v_j = target/(t+TINY), col_err = |v_j*t - target|
// mode 1: out_j = vin_j * target / max(t*vin_j, TINY)   (final column factor)
__global__ __launch_bounds__(256) void colCombine(
    const float* __restrict__ tpart, const float* __restrict__ vin,
    float* __restrict__ out, Ctrl* ctrl, int mode, int guard) {
  if (guard && ctrl->done) return;
  const int j = blockIdx.x * 256 + threadIdx.x;
  if (j >= K_N) return;
  float t = 0.0f;
#pragma unroll
  for (int s = 0; s < ISPLIT; ++s) t += tpart[s * K_PAD + j];
  if (mode == 0) {
    const float nv = kTargetC / (t + kTiny);
    out[j] = nv;
    const float e = fabsf(nv * t - kTargetC);
    atomicMax(&ctrl->col_err, __float_as_uint(e));
  } else {
    const float vj = vin[j];
    out[j] = vj * (kTargetC / fmaxf(t * vj, kTiny));
  }
}

__global__ void sinkCheck(Ctrl* c) {
  if (c->done) return;
  c->it++;
  const float err = fmaxf(__uint_as_float(c->row_err), __uint_as_float(c->col_err));
  if (c->it >= kMinIters && (c->it >= kMaxIters || err <= kTol)) c->done = 1;
  c->row_err = 0u;
  c->col_err = 0u;
}

// Per-row entropy term and max of P_ij = arow_i * Q_ij * bcol_j.
__global__ __launch_bounds__(256) void statsPass(
    const float* __restrict__ Q, const float* __restrict__ arow,
    const float* __restrict__ bcol, float* __restrict__ entArr,
    float* __restrict__ maxArr) {
  __shared__ float red[256];
  const int row = blockIdx.x;
  const float a = arow[row];
  const float* q = Q + (size_t)row * K_N;
  float se = 0.0f, mxv = -3.4e38f;
  for (int j = threadIdx.x; j < K_N; j += 256) {
    const float pv = a * q[j] * bcol[j];
    se += pv * logf(pv + kTiny);
    mxv = fmaxf(mxv, pv);
  }
  const float seT = blkSum(se, red);
  const float mxT = blkMax(mxv, red);
  if (threadIdx.x == 0) { entArr[row] = seT; maxArr[row] = mxT; }
}

// Fused per-(row, crop): m = max(x/T), lse, and dots with Pa/Pb rows.
// sum_j P*logp = arow*sum_j(Q*bcol*x) - (m + lse)  since P rows sum to 1.
__global__ __launch_bounds__(256) void lossPass(
    Ptrs8 ptrs, const float* __restrict__ Qa, const float* __restrict__ Qb,
    const float* __restrict__ arowA, const float* __restrict__ bcolA,
    const float* __restrict__ arowB, const float* __restrict__ bcolB,
    float* __restrict__ lossArr) {
  __shared__ float xs[K_PAD];
  __shared__ float red[256];
  const int row = blockIdx.x;
  const int c = blockIdx.y;
  const float* x = ptrs.p[c] + (size_t)row * K_N;
  float m = -3.4e38f;
  for (int j = threadIdx.x; j < K_N; j += 256) {
    const float v = x[j] * (1.0f / kTemp);
    xs[j] = v;
    m = fmaxf(m, v);
  }
  m = blkMax(m, red);
  float se = 0.0f;
  for (int j = threadIdx.x; j < K_N; j += 256) se += expf(xs[j] - m);
  const float l = logf(blkSum(se, red));
  const float* qa = Qa + (size_t)row * K_N;
  const float* qb = Qb + (size_t)row * K_N;
  float sa = 0.0f, sb = 0.0f;
  for (int j = threadIdx.x; j < K_N; j += 256) {
    const float xv = xs[j];
    sa += qa[j] * bcolA[j] * xv;
    sb += qb[j] * bcolB[j] * xv;
  }
  const float saT = blkSum(sa, red);
  const float sbT = blkSum(sb, red);
  if (threadIdx.x == 0) {
    lossArr[(size_t)c * B_N + row] = arowA[row] * saT - (m + l);
    lossArr[(size_t)(C_N + c) * B_N + row] = arowB[row] * sbT - (m + l);
  }
}

__global__ __launch_bounds__(256) void finalizeK(
    const float* __restrict__ entA, const float* __restrict__ maxA,
    const float* __restrict__ entB, const float* __restrict__ maxB,
    const float* __restrict__ lossArr, const int* __restrict__ idxa,
    const int* __restrict__ idxb, float* __restrict__ out) {
  __shared__ float red[256];
  __shared__ float res[20];
  const float* arrs[4] = {entA, maxA, entB, maxB};
  for (int a = 0; a < 4; ++a) {
    float s = 0.0f;
    for (int i = threadIdx.x; i < B_N; i += 256) s += arrs[a][i];
    s = blkSum(s, red);
    if (threadIdx.x == 0) res[a] = s;
  }
  for (int q = 0; q < 16; ++q) {
    float s = 0.0f;
    const float* pq = lossArr + (size_t)q * B_N;
    for (int i = threadIdx.x; i < B_N; i += 256) s += pq[i];
    s = blkSum(s, red);
    if (threadIdx.x == 0) res[4 + q] = s;
  }
  __syncthreads();
  if (threadIdx.x == 0) {
    const float invB = 1.0f / (float)B_N;
    const float ent = 0.5f * ((-res[0] * invB) + (-res[2] * invB));
    const float qm = 0.5f * ((res[1] * invB) + (res[3] * invB));
    const int ia = *idxa, ib = *idxb;
    float acc = 0.0f;
    int cnt = 0;
    for (int c = 0; c < C_N; ++c) {
      if (c != ia) { acc += -(res[4 + c] * invB); cnt++; }
      if (c != ib) { acc += -(res[4 + 8 + c] * invB); cnt++; }
    }
    out[0] = acc / (float)cnt;
    out[1] = ent;
    out[2] = qm;
  }
}

extern "C" void kernel_launch(void* const* d_in, const int* in_sizes, int n_in,
                              void* d_out, int out_size, void* d_ws, size_t ws_size,
                              hipStream_t stream) {
  (void)in_sizes; (void)n_in; (void)out_size; (void)ws_size;
  Ptrs8 ptrs;
  for (int i = 0; i < 8; ++i) ptrs.p[i] = (const float*)d_in[i];
  const int* idxa = (const int*)d_in[8];
  const int* idxb = (const int*)d_in[9];
  float* outp = (float*)d_out;

  float* ws = (float*)d_ws;
  const size_t BK = (size_t)B_N * K_N;
  float* Qs[2];
  Qs[0] = ws;
  Qs[1] = ws + BK;
  float* p = ws + 2 * BK;
  float *u[2], *w[2], *arow[2], *v[2], *bcol[2], *tpart[2], *rpart[2], *ent[2], *mx[2];
  for (int s = 0; s < 2; ++s) {
    u[s] = p; p += B_N;
    w[s] = p; p += B_N;
    arow[s] = p; p += B_N;
    v[s] = p; p += K_PAD;
    bcol[s] = p; p += K_PAD;
    tpart[s] = p; p += ISPLIT * K_PAD;
    rpart[s] = p; p += KSPLIT * B_N;
    ent[s] = p; p += B_N;
    mx[s] = p; p += B_N;
  }
  float* lossArr = p; p += 2 * C_N * B_N;
  Ctrl* ctrl[2];
  ctrl[0] = (Ctrl*)p;
  ctrl[1] = ctrl[0] + 1;

  const dim3 blk(256);
  const dim3 rowGrid(B_N / 128, KSPLIT);
  const dim3 rowGrid1((B_N + 255) / 256);
  const dim3 colGrid((K_N + 255) / 256, ISPLIT);
  const dim3 colGrid1((K_N + 255) / 256);

  buildQ<<<B_N, blk, 0, stream>>>(ptrs, idxa, Qs[0]);
  buildQ<<<B_N, blk, 0, stream>>>(ptrs, idxb, Qs[1]);
  for (int s = 0; s < 2; ++s)
    sinkInit<<<16, blk, 0, stream>>>(u[s], v[s], ctrl[s]);

  // Sinkhorn: 100 recorded iterations; kernels no-op once ctrl->done is set.
  for (int it = 0; it < kMaxIters; ++it) {
    for (int s = 0; s < 2; ++s) {
      rowPass<<<rowGrid, blk, 0, stream>>>(Qs[s], v[s], rpart[s], ctrl[s], 1);
      rowCombine<<<rowGrid1, blk, 0, stream>>>(rpart[s], nullptr, u[s], ctrl[s], 0, 1);
      colPass<<<colGrid, blk, 0, stream>>>(Qs[s], u[s], tpart[s], ctrl[s], 1);
      colCombine<<<colGrid1, blk, 0, stream>>>(tpart[s], nullptr, v[s], ctrl[s], 0, 1);
      rowPass<<<rowGrid, blk, 0, stream>>>(Qs[s], v[s], rpart[s], ctrl[s], 1);
      rowCombine<<<rowGrid1, blk, 0, stream>>>(rpart[s], u[s], nullptr, ctrl[s], 1, 1);
      sinkCheck<<<1, 1, 0, stream>>>(ctrl[s]);
    }
  }

  // Final row/col/row renormalization folded into factors: P = arow_i*Q_ij*bcol_j.
  for (int s = 0; s < 2; ++s) {
    rowPass<<<rowGrid, blk, 0, stream>>>(Qs[s], v[s], rpart[s], ctrl[s], 0);
    rowCombine<<<rowGrid1, blk, 0, stream>>>(rpart[s], u[s], w[s], ctrl[s], 2, 0);
    colPass<<<colGrid, blk, 0, stream>>>(Qs[s], w[s], tpart[s], ctrl[s], 0);
    colCombine<<<colGrid1, blk, 0, stream>>>(tpart[s], v[s], bcol[s], ctrl[s], 1, 0);
    rowPass<<<rowGrid, blk, 0, stream>>>(Qs[s], bcol[s], rpart[s], ctrl[s], 0);
    rowCombine<<<rowGrid1, blk, 0, stream>>>(rpart[s], w[s], arow[s], ctrl[s], 2, 0);
    statsPass<<<B_N, blk, 0, stream>>>(Qs[s], arow[s], bcol[s], ent[s], mx[s]);
  }

  lossPass<<<dim3(B_N, C_N), blk, 0, stream>>>(ptrs, Qs[0], Qs[1], arow[0], bcol[0],
                                               arow[1], bcol[1], lossArr);
  finalizeK<<<1, blk, 0, stream>>>(ent[0], mx[0], ent[1], mx[1], lossArr, idxa, idxb,
                                   outp);
}